// DynamicMRConv4d_65180423685242
// MI455X (gfx1250) — compile-verified
//
#include <hip/hip_runtime.h>
#include <math.h>

#define B_   32
#define C_   192
#define H_   64
#define W_   64
#define OC_  192
#define HW_  (H_ * W_)
#define K2C_ (2 * C_)

typedef __attribute__((ext_vector_type(2))) float v2f;
typedef __attribute__((ext_vector_type(8))) float v8f;

// ---------------------------------------------------------------------------
// Kernel 1: per-batch threshold = mean(norm) - std(norm, ddof=1)
// norm[p] = sum_c |x[b,c,p] - x[b,c,roll(p, 32,32)]|
// ---------------------------------------------------------------------------
__global__ void thresh_kernel(const float* __restrict__ x,
                              float* __restrict__ thresh) {
    __shared__ float s1[256], s2[256];
    const int b = blockIdx.x;
    const float* xb = x + (size_t)b * C_ * HW_;

    float a1 = 0.f, a2 = 0.f;
    for (int p = threadIdx.x; p < HW_; p += 256) {
        const int h = p >> 6, w = p & 63;
        const int pe = (((h + 32) & 63) << 6) | ((w + 32) & 63);
        float n = 0.f;
        for (int c = 0; c < C_; ++c)
            n += fabsf(xb[(size_t)c * HW_ + p] - xb[(size_t)c * HW_ + pe]);
        a1 += n;
        a2 += n * n;
    }
    s1[threadIdx.x] = a1;
    s2[threadIdx.x] = a2;
    __syncthreads();
    for (int s = 128; s > 0; s >>= 1) {
        if (threadIdx.x < s) {
            s1[threadIdx.x] += s1[threadIdx.x + s];
            s2[threadIdx.x] += s2[threadIdx.x + s];
        }
        __syncthreads();
    }
    if (threadIdx.x == 0) {
        const float n = (float)HW_;
        const float mean = s1[0] / n;
        const float var = (s2[0] - s1[0] * s1[0] / n) / (n - 1.f);  // ddof=1
        thresh[b] = mean - sqrtf(fmaxf(var, 0.f));
    }
}

// ---------------------------------------------------------------------------
// Kernel 2: x_j accumulation. One workgroup per (b,h) row.
// LDS: X row tile (192x64), XJ accumulator (192x64), XR streamed row (192x64),
//      Dp partial-dist scratch (4x64). ~148 KB (CDNA5 WGP has 320 KB LDS).
// ---------------------------------------------------------------------------
__global__ void xj_kernel(const float* __restrict__ x,
                          const float* __restrict__ thresh,
                          float* __restrict__ xj) {
    extern __shared__ float smem[];
    float* Xs = smem;                 // 12288
    float* XJ = Xs + C_ * W_;         // 12288
    float* XR = XJ + C_ * W_;         // 12288
    float* Dp = XR + C_ * W_;         // 256 = [4][64]

    const int b = blockIdx.x >> 6;
    const int h = blockIdx.x & 63;
    const int tid = threadIdx.x;
    const int w  = tid & 63;   // spatial column owned by this thread
    const int cp = tid >> 6;   // channel-partition 0..3 (48 channels each)
    const int c0 = cp * (C_ / 4), c1 = c0 + (C_ / 4);

    const float* xb = x + (size_t)b * C_ * HW_;

    // load X row tile, zero XJ (coalesced: 64 consecutive w per channel row)
    for (int f = tid; f < C_ * W_; f += 256) {
        const int c = f >> 6, ww = f & 63;
        Xs[f] = xb[(size_t)c * HW_ + h * W_ + ww];
        XJ[f] = 0.f;
    }
    __syncthreads();

    const float th = thresh[b];

    // ---- W-direction shifts: all data already in LDS ----
    for (int j = 8; j < W_; j += 8) {
        const int wr = (w + j) & 63;
        float d = 0.f;
        for (int c = c0; c < c1; ++c)
            d += fabsf(Xs[c * 64 + w] - Xs[c * 64 + wr]);
        Dp[cp * 64 + w] = d;
        __syncthreads();
        const float dist = Dp[w] + Dp[64 + w] + Dp[128 + w] + Dp[192 + w];
        if (dist < th) {  // mask==0 is a no-op since XJ >= 0 under running max
            for (int c = c0; c < c1; ++c) {
                const float diff = Xs[c * 64 + wr] - Xs[c * 64 + w];
                XJ[c * 64 + w] = fmaxf(XJ[c * 64 + w], diff);
            }
        }
        __syncthreads();
    }

    // ---- H-direction shifts: stream row (h+i)%64 through XR ----
    for (int i = 8; i < H_; i += 8) {
        const int h2 = (h + i) & 63;
        for (int f = tid; f < C_ * W_; f += 256) {
            const int c = f >> 6, ww = f & 63;
            XR[f] = xb[(size_t)c * HW_ + h2 * W_ + ww];
        }
        __syncthreads();
        float d = 0.f;
        for (int c = c0; c < c1; ++c)
            d += fabsf(Xs[c * 64 + w] - XR[c * 64 + w]);
        Dp[cp * 64 + w] = d;
        __syncthreads();
        const float dist = Dp[w] + Dp[64 + w] + Dp[128 + w] + Dp[192 + w];
        if (dist < th) {
            for (int c = c0; c < c1; ++c) {
                const float diff = XR[c * 64 + w] - Xs[c * 64 + w];
                XJ[c * 64 + w] = fmaxf(XJ[c * 64 + w], diff);
            }
        }
        __syncthreads();
    }

    // write XJ row tile out
    float* xjb = xj + (size_t)b * C_ * HW_;
    for (int f = tid; f < C_ * W_; f += 256) {
        const int c = f >> 6, ww = f & 63;
        xjb[(size_t)c * HW_ + h * W_ + ww] = XJ[f];
    }
}

// ---------------------------------------------------------------------------
// Kernel 3: 1x1 conv as WMMA fp32 GEMM.
// Block = 384 threads (12 waves), one block per (b, 64-position row tile).
// x_cat panel [384 k][64 n] staged once in 96 KB LDS, stored K-PAIR
// INTERLEAVED:  Bs[(k/2)*128 + n*2 + (k&1)]  so a lane's {k,k+1} B-fragment
// is one aligned ds_load_b64 (no repacking movs before v_wmma).
// Wave w owns oc-tile w: a 16x64 strip = 4 accumulators, K swept with
// V_WMMA_F32_16X16X4_F32 (one A b64 load feeds 4 WMMAs).
// ---------------------------------------------------------------------------
__global__ void gemm_wmma_kernel(const float* __restrict__ x,
                                 const float* __restrict__ xj,
                                 const float* __restrict__ conv_w,
                                 const float* __restrict__ conv_b,
                                 float* __restrict__ y) {
    extern __shared__ float Bs[];          // 384*64 fp32 = 96 KB, interleaved

    const int b  = blockIdx.x >> 6;        // batch
    const int pt = blockIdx.x & 63;        // 64-wide position tile (one h row)
    const int p0 = pt * 64;

    const int wave = threadIdx.x >> 5;     // 0..11 -> oc tile
    const int lane = threadIdx.x & 31;
    const int oc0  = wave * 16;

    // cooperative stage: two channel rows per step, interleave k-pairs,
    // b128 global loads + b128 LDS stores
    float4* Bs4 = (float4*)Bs;
    for (int f = threadIdx.x; f < (K2C_ / 2) * 16; f += 384) {   // 3072
        const int cp = f >> 4;             // k-pair index 0..191
        const int q  = f & 15;             // float4 column group (n = 4q..4q+3)
        const float* s0;
        const float* s1;
        if (cp < C_ / 2) {                 // both rows in x
            s0 = x + ((size_t)b * C_ + 2 * cp) * HW_;
            s1 = s0 + HW_;
        } else {                           // both rows in x_j
            s0 = xj + ((size_t)b * C_ + (2 * cp - C_)) * HW_;
            s1 = s0 + HW_;
        }
        const float4 a4 = *(const float4*)(s0 + p0 + q * 4);
        const float4 b4 = *(const float4*)(s1 + p0 + q * 4);
        float4 lo, hi;
        lo.x = a4.x; lo.y = b4.x; lo.z = a4.y; lo.w = b4.y;
        hi.x = a4.z; hi.y = b4.z; hi.z = a4.w; hi.w = b4.w;
        Bs4[cp * 32 + q * 2]     = lo;
        Bs4[cp * 32 + q * 2 + 1] = hi;
    }
    __syncthreads();

    // A-fragment (16x4 f32): lanes 0-15 -> k+0,k+1 ; lanes 16-31 -> k+2,k+3
    const int m  = lane & 15;
    const int kh = lane >> 4;
    const float* arow = conv_w + (size_t)(oc0 + m) * K2C_ + 2 * kh;
    const float2* Bs2 = (const float2*)Bs; // float2 index = kpair*64 + n

    v8f acc0 = {}, acc1 = {}, acc2 = {}, acc3 = {};
#pragma unroll 4
    for (int k = 0; k < K2C_; k += 4) {
        const float2 av = *(const float2*)(arow + k);   // 8B-aligned b64 load
        v2f a;  a[0] = av.x;  a[1] = av.y;
        const float2* bp = Bs2 + ((k >> 1) + kh) * 64 + m;
        const float2 f0 = bp[0], f1 = bp[16], f2 = bp[32], f3 = bp[48];
        v2f b0; b0[0] = f0.x; b0[1] = f0.y;
        v2f b1; b1[0] = f1.x; b1[1] = f1.y;
        v2f b2; b2[0] = f2.x; b2[1] = f2.y;
        v2f b3; b3[0] = f3.x; b3[1] = f3.y;
        acc0 = __builtin_amdgcn_wmma_f32_16x16x4_f32(false, a, false, b0,
                                                     (short)0, acc0, false, false);
        acc1 = __builtin_amdgcn_wmma_f32_16x16x4_f32(false, a, false, b1,
                                                     (short)0, acc1, false, false);
        acc2 = __builtin_amdgcn_wmma_f32_16x16x4_f32(false, a, false, b2,
                                                     (short)0, acc2, false, false);
        acc3 = __builtin_amdgcn_wmma_f32_16x16x4_f32(false, a, false, b3,
                                                     (short)0, acc3, false, false);
    }

    // C/D layout: VGPR r -> row r (+8 for lanes>=16), column = lane&15
    float* ybase = y + (size_t)b * OC_ * HW_ + p0;
#pragma unroll
    for (int r = 0; r < 8; ++r) {
        const int oc = oc0 + r + 8 * kh;
        const float bias = conv_b[oc];
        float* yrow = ybase + (size_t)oc * HW_ + m;
        yrow[0]  = acc0[r] + bias;
        yrow[16] = acc1[r] + bias;
        yrow[32] = acc2[r] + bias;
        yrow[48] = acc3[r] + bias;
    }
}

// ---------------------------------------------------------------------------
// Kernel 4: per-channel batch-norm statistics (biased var over B*H*W)
// ---------------------------------------------------------------------------
__global__ void bnstat_kernel(const float* __restrict__ y,
                              float* __restrict__ bn_mean,
                              float* __restrict__ bn_rstd) {
    __shared__ float s1[256], s2[256];
    const int o = blockIdx.x;
    float a1 = 0.f, a2 = 0.f;
    for (int idx = threadIdx.x; idx < B_ * HW_; idx += 256) {
        const int b = idx >> 12;
        const int p = idx & (HW_ - 1);
        const float v = y[((size_t)b * OC_ + o) * HW_ + p];
        a1 += v;
        a2 += v * v;
    }
    s1[threadIdx.x] = a1;
    s2[threadIdx.x] = a2;
    __syncthreads();
    for (int s = 128; s > 0; s >>= 1) {
        if (threadIdx.x < s) {
            s1[threadIdx.x] += s1[threadIdx.x + s];
            s2[threadIdx.x] += s2[threadIdx.x + s];
        }
        __syncthreads();
    }
    if (threadIdx.x == 0) {
        const float n = (float)(B_ * HW_);
        const float mean = s1[0] / n;
        const float var = s2[0] / n - mean * mean;  // biased
        bn_mean[o] = mean;
        bn_rstd[o] = rsqrtf(var + 1e-5f);
    }
}

// ---------------------------------------------------------------------------
// Kernel 5: in-place BN affine + exact GELU, float4-vectorized
// ---------------------------------------------------------------------------
__global__ void bngelu_kernel(float* __restrict__ y,
                              const float* __restrict__ bn_mean,
                              const float* __restrict__ bn_rstd,
                              const float* __restrict__ gamma,
                              const float* __restrict__ beta) {
    const size_t idx = (size_t)blockIdx.x * blockDim.x + threadIdx.x;
    const size_t n4 = (size_t)B_ * OC_ * HW_ / 4;
    if (idx >= n4) return;
    const size_t e = idx * 4;
    const int o = (int)((e / HW_) % OC_);  // HW_ % 4 == 0 -> float4 stays in-channel

    const float mu = bn_mean[o], rs = bn_rstd[o];
    const float g = gamma[o], bt = beta[o];

    float4 v = ((float4*)y)[idx];
    float* vp = (float*)&v;
#pragma unroll
    for (int i = 0; i < 4; ++i) {
        const float u = (vp[i] - mu) * rs * g + bt;
        vp[i] = 0.5f * u * (1.f + erff(u * 0.70710678118654752f));
    }
    ((float4*)y)[idx] = v;
}

// ---------------------------------------------------------------------------
// Launch
// ---------------------------------------------------------------------------
extern "C" void kernel_launch(void* const* d_in, const int* in_sizes, int n_in,
                              void* d_out, int out_size, void* d_ws, size_t ws_size,
                              hipStream_t stream) {
    const float* x      = (const float*)d_in[0];
    const float* conv_w = (const float*)d_in[1];
    const float* conv_b = (const float*)d_in[2];
    const float* gamma  = (const float*)d_in[3];
    const float* beta   = (const float*)d_in[4];
    // d_in[5] = K (==8), baked into the shift loops

    float* out = (float*)d_out;
    float* ws  = (float*)d_ws;
    float* thresh  = ws;         // 32 floats
    float* bn_mean = ws + 32;    // 192 floats
    float* bn_rstd = ws + 224;   // 192 floats
    float* xj      = ws + 512;   // B*C*H*W floats (100.7 MB)

    // 1. per-batch threshold
    thresh_kernel<<<B_, 256, 0, stream>>>(x, thresh);

    // 2. x_j via LDS-resident row tiles (~148 KB dynamic LDS per workgroup)
    const size_t smem_xj = (size_t)(3 * C_ * W_ + 256) * sizeof(float);
    xj_kernel<<<B_ * H_, 256, smem_xj, stream>>>(x, thresh, xj);

    // 3. WMMA fp32 GEMM -> y directly into d_out (96 KB LDS x_cat panel)
    const size_t smem_gemm = (size_t)K2C_ * 64 * sizeof(float);
    gemm_wmma_kernel<<<B_ * 64, 384, smem_gemm, stream>>>(x, xj, conv_w, conv_b, out);

    // 4. BN statistics
    bnstat_kernel<<<OC_, 256, 0, stream>>>(out, bn_mean, bn_rstd);

    // 5. BN affine + exact GELU in place
    const size_t n4 = (size_t)B_ * OC_ * HW_ / 4;
    bngelu_kernel<<<(unsigned)((n4 + 255) / 256), 256, 0, stream>>>(
        out, bn_mean, bn_rstd, gamma, beta);
}